// GIN_10574209483251
// MI455X (gfx1250) — compile-verified
//
#include <hip/hip_runtime.h>

typedef __attribute__((ext_vector_type(2))) float v2f;
typedef __attribute__((ext_vector_type(8))) float v8f;

#define DH 128       // D_IN == D_HID == 128
#define NG 512       // N_GRAPHS
#define DOUT 10      // D_OUT
#define A_STRIDE 132 // padded LDS stride (floats): keeps 8B alignment, avoids bank conflicts

// Native non-returning f32 atomic add, device scope (L2 atomic units).
// Avoids any CAS-loop expansion of fp atomicrmw.
__device__ __forceinline__ void atomic_add_f32(float* p, float v) {
    asm volatile("global_atomic_add_f32 %0, %1, off scope:SCOPE_DEV"
                 :: "v"(p), "v"(v)
                 : "memory");
}

// Raw LDS byte offset (addrspace(3) value) for async-to-LDS instructions.
__device__ __forceinline__ unsigned lds_off(const void* p) {
    return (unsigned)(unsigned long long)(__attribute__((address_space(3))) const void*)p;
}

// ---------------------------------------------------------------------------
// Edge aggregation: AGG[dst] += X[src].  32 threads per edge, float4 each.
// X and AGG are 51.2MB each -> resident in 192MB L2; atomics resolve at L2.
// ---------------------------------------------------------------------------
__global__ __launch_bounds__(256) void gin_agg_kernel(
    const float* __restrict__ X, const long long* __restrict__ src,
    const long long* __restrict__ dst, float* __restrict__ AGG, int n_edges)
{
    int t = blockIdx.x * 256 + threadIdx.x;
    int e = t >> 5;
    if (e >= n_edges) return;
    int c = (t & 31) << 2;
    int s = (int)src[e];
    int d = (int)dst[e];
    const float4 v = *(const float4*)(X + (size_t)s * DH + c);
    float* p = AGG + (size_t)d * DH + c;
    atomic_add_f32(p + 0, v.x);
    atomic_add_f32(p + 1, v.y);
    atomic_add_f32(p + 2, v.z);
    atomic_add_f32(p + 3, v.w);
}

// ---------------------------------------------------------------------------
// Fused OUT = act((X + AGG) @ W^T + bias) with V_WMMA_F32_16X16X4_F32.
// Block = 256 threads = 8 waves; wave w owns N-tile w and keeps all 32
// K-step B fragments of W in registers for the whole block (grid-stride
// over M-tiles). Per M-tile the X and AGG 16x128 tiles are DMA'd into LDS
// with GLOBAL_LOAD_ASYNC_TO_LDS_B128 (ASYNCcnt), and the GIN "x + agg" add
// is folded into the per-fragment LDS read feeding the WMMA.
//
// f32 WMMA operand layout (ISA 7.12.2):
//   A (16x4):  lane&15 = M row; (lane>>4)*2 = K base; VGPR0/1 = K, K+1
//   B (4x16):  lane&15 = N col; (lane>>4)*2 = K base; VGPR0/1 = K, K+1
//   C/D:       VGPR r -> M = r (lanes 0-15) / r+8 (lanes 16-31), N = lane&15
// ---------------------------------------------------------------------------
template <bool RELU>
__global__ __launch_bounds__(256) void gin_gemm_kernel(
    const float* __restrict__ X, const float* __restrict__ AGG,
    const float* __restrict__ W, const float* __restrict__ Bias,
    float* __restrict__ OUT, int n_rows)
{
    __shared__ float Xt[16 * A_STRIDE];
    __shared__ float At[16 * A_STRIDE];
    const int lane  = threadIdx.x & 31;
    const int wave  = threadIdx.x >> 5;
    const int ntile = wave << 4;
    const int idx16 = lane & 15;
    const int kb    = (lane >> 4) << 1;

    // Hoist B fragments: B[k][n] = W[n][k]  (OUT = A @ W^T)
    v2f bfrag[32];
#pragma unroll
    for (int ks = 0; ks < 32; ++ks) {
        const float* wp = W + (size_t)(ntile + idx16) * DH + (ks << 2) + kb;
        bfrag[ks].x = wp[0];
        bfrag[ks].y = wp[1];
    }

    const int n_mtiles = (n_rows + 15) >> 4;
    for (int mt = blockIdx.x; mt < n_mtiles; mt += gridDim.x) {
        const int mtile = mt << 4;

        __syncthreads(); // previous tile's LDS reads complete
        for (int i = threadIdx.x; i < (16 * DH) / 4; i += 256) {
            int r = i >> 5;          // row within tile
            int c = (i & 31) << 2;   // col (float4 granularity)
            int row = mtile + r;
            float* lx = Xt + r * A_STRIDE + c;
            float* la = At + r * A_STRIDE + c;
            if (row < n_rows) {
                const float* gx = X + (size_t)row * DH + c;
                const float* ga = AGG + (size_t)row * DH + c;
                unsigned ox = lds_off(lx);
                unsigned oa = lds_off(la);
                asm volatile("global_load_async_to_lds_b128 %0, %1, off"
                             :: "v"(ox), "v"(gx) : "memory");
                asm volatile("global_load_async_to_lds_b128 %0, %1, off"
                             :: "v"(oa), "v"(ga) : "memory");
            } else {
                *(float4*)lx = make_float4(0.f, 0.f, 0.f, 0.f);
                *(float4*)la = make_float4(0.f, 0.f, 0.f, 0.f);
            }
        }
        asm volatile("s_wait_asynccnt 0x0" ::: "memory"); // DMA landed in LDS
        __syncthreads();

        v8f acc = {};
#pragma unroll
        for (int ks = 0; ks < 32; ++ks) {
            const int o = idx16 * A_STRIDE + (ks << 2) + kb;
            v2f a;
            a.x = Xt[o + 0] + At[o + 0];
            a.y = Xt[o + 1] + At[o + 1];
            acc = __builtin_amdgcn_wmma_f32_16x16x4_f32(
                false, a, false, bfrag[ks], (short)0, acc, false, false);
        }

        const int   ncol  = ntile + idx16;
        const float bias  = Bias[ncol];
        const int   mbase = mtile + ((lane >> 4) << 3);
#pragma unroll
        for (int r = 0; r < 8; ++r) {
            int row = mbase + r;
            if (row < n_rows) {
                float v = acc[r] + bias;
                if (RELU) v = fmaxf(v, 0.f);
                OUT[(size_t)row * DH + ncol] = v;
            }
        }
    }
}

// ---------------------------------------------------------------------------
// Mean-pool part 1: sums[batch[n]] += H[n]; counts[batch[n]] += 1
// ---------------------------------------------------------------------------
__global__ __launch_bounds__(256) void gin_pool_kernel(
    const float* __restrict__ H, const long long* __restrict__ batch,
    float* __restrict__ sums, float* __restrict__ counts, int n_nodes)
{
    int t = blockIdx.x * 256 + threadIdx.x;
    int node = t >> 5;
    if (node >= n_nodes) return;
    int c = (t & 31) << 2;
    int g = (int)batch[node];
    const float4 v = *(const float4*)(H + (size_t)node * DH + c);
    float* p = sums + (size_t)g * DH + c;
    atomic_add_f32(p + 0, v.x);
    atomic_add_f32(p + 1, v.y);
    atomic_add_f32(p + 2, v.z);
    atomic_add_f32(p + 3, v.w);
    if ((t & 31) == 0) atomic_add_f32(counts + g, 1.0f);
}

// ---------------------------------------------------------------------------
// Head: out[g][o] = dot(sums[g], Wg[o]) / max(count[g],1) + bg[o]   (512x10)
// ---------------------------------------------------------------------------
__global__ __launch_bounds__(256) void gin_head_kernel(
    const float* __restrict__ sums, const float* __restrict__ counts,
    const float* __restrict__ Wg, const float* __restrict__ bg,
    float* __restrict__ out)
{
    int t = blockIdx.x * 256 + threadIdx.x;
    if (t >= NG * DOUT) return;
    int g = t / DOUT, o = t % DOUT;
    float inv = 1.0f / fmaxf(counts[g], 1.0f);
    float acc = 0.f;
    const float* sp = sums + (size_t)g * DH;
    const float* wp = Wg + (size_t)o * DH;
#pragma unroll 8
    for (int d = 0; d < DH; ++d) acc += sp[d] * wp[d];
    out[t] = acc * inv + bg[o];
}

// ---------------------------------------------------------------------------
extern "C" void kernel_launch(void* const* d_in, const int* in_sizes, int n_in,
                              void* d_out, int out_size, void* d_ws, size_t ws_size,
                              hipStream_t stream)
{
    const float*     x     = (const float*)d_in[0];
    const long long* ei    = (const long long*)d_in[1]; // int64 [2, E]
    const long long* batch = (const long long*)d_in[2]; // int64 [N]
    const float* W0 = (const float*)d_in[3];
    const float* b0 = (const float*)d_in[4];
    const float* W1 = (const float*)d_in[5];
    const float* b1 = (const float*)d_in[6];
    const float* W2 = (const float*)d_in[7];
    const float* b2 = (const float*)d_in[8];
    const float* Wg = (const float*)d_in[9];
    const float* bg = (const float*)d_in[10];

    const int n_nodes = in_sizes[0] / DH;
    const int n_edges = in_sizes[1] / 2;
    const long long* src = ei;
    const long long* dst = ei + n_edges;

    float* ws = (float*)d_ws;
    const size_t NF = (size_t)n_nodes * DH;
    float* agg    = ws;               // [N,128]
    float* hA     = ws + NF;          // [N,128]
    float* hB     = ws + 2 * NF;      // [N,128]
    float* sums   = ws + 3 * NF;      // [512,128]
    float* counts = sums + (size_t)NG * DH; // [512]

    const int agg_blocks  = (n_edges * 32 + 255) / 256;
    const int gemm_blocks = 640;
    const int pool_blocks = (n_nodes * 32 + 255) / 256;

    // Layer 0: h0 = relu((x + agg(x)) @ W0^T + b0)
    hipMemsetAsync(agg, 0, NF * sizeof(float), stream);
    gin_agg_kernel<<<agg_blocks, 256, 0, stream>>>(x, src, dst, agg, n_edges);
    gin_gemm_kernel<true><<<gemm_blocks, 256, 0, stream>>>(x, agg, W0, b0, hA, n_nodes);

    // Layer 1
    hipMemsetAsync(agg, 0, NF * sizeof(float), stream);
    gin_agg_kernel<<<agg_blocks, 256, 0, stream>>>(hA, src, dst, agg, n_edges);
    gin_gemm_kernel<true><<<gemm_blocks, 256, 0, stream>>>(hA, agg, W1, b1, hB, n_nodes);

    // Layer 2 (no relu)
    hipMemsetAsync(agg, 0, NF * sizeof(float), stream);
    gin_agg_kernel<<<agg_blocks, 256, 0, stream>>>(hB, src, dst, agg, n_edges);
    gin_gemm_kernel<false><<<gemm_blocks, 256, 0, stream>>>(hB, agg, W2, b2, hA, n_nodes);

    // Mean pool + classifier head
    hipMemsetAsync(sums, 0, ((size_t)NG * DH + NG) * sizeof(float), stream);
    gin_pool_kernel<<<pool_blocks, 256, 0, stream>>>(hA, batch, sums, counts, n_nodes);
    gin_head_kernel<<<(NG * DOUT + 255) / 256, 256, 0, stream>>>(
        sums, counts, Wg, bg, (float*)d_out);
}